// AmpNorm_3590592659803
// MI455X (gfx1250) — compile-verified
//
#include <hip/hip_runtime.h>
#include <hip/hip_bf16.h>
#include <math.h>

// Problem geometry: x [32,3,384,384] f32, running_amp [3,384,384] f32.
#define NDIM 384
#define NN (384 * 384)                   // elements per plane
#define NPLANES 96                       // 32 * 3
#define MTILE 32                         // macro tile (per 128-thread WG)
#define NMT (NDIM / MTILE)               // 12
#define TILES_PER_PLANE (NMT * NMT)      // 144
#define KC 32                            // K chunk staged in LDS per step
#define NCHUNK (NDIM / KC)               // 12
#define MOMENTUM 0.1f

typedef float v2f __attribute__((ext_vector_type(2)));
typedef float v8f __attribute__((ext_vector_type(8)));
typedef unsigned int u32x4 __attribute__((ext_vector_type(4)));
typedef int i32x4 __attribute__((ext_vector_type(4)));
typedef int i32x8 __attribute__((ext_vector_type(8)));

// ---------------------------------------------------------------------------
// WMMA helper: D = A(16x4 f32) * B(4x16 f32) + C(16x16 f32)
// ---------------------------------------------------------------------------
__device__ __forceinline__ v8f wmma4(v2f a, v2f b, v8f c) {
  return __builtin_amdgcn_wmma_f32_16x16x4_f32(
      false, a, false, b, (short)0, c, false, false);
}

// C/D store (16x16 f32): VGPR i holds row (i + 8*(lane>>4)), col = lane&15.
__device__ __forceinline__ void store_c(float* __restrict__ C, int rowBase,
                                        int colBase, int lane, v8f c,
                                        float scale) {
  int n = lane & 15;
  int hi = lane >> 4;
#pragma unroll
  for (int i = 0; i < 8; ++i) {
    C[(size_t)(rowBase + i + 8 * hi) * NDIM + colBase + n] = c[i] * scale;
  }
}

// ---------------------------------------------------------------------------
// Tensor Data Mover: async 32x32 f32 tile (row-major, ld=384) -> LDS.
// D# layout per CDNA5 ISA (group0 128b, group1 256b).
// 6-arg builtin form (clang-23 / therock-10.0 headers).
// ---------------------------------------------------------------------------
__device__ __forceinline__ void tdm_load_tile32(const float* gsrc,
                                                unsigned ldsAddr) {
  unsigned long long ga = (unsigned long long)(uintptr_t)gsrc;
  u32x4 g0;
  g0[0] = 1u;                                   // count=1 (valid), user mode
  g0[1] = ldsAddr;                              // bits 63:32  lds_addr (bytes)
  g0[2] = (unsigned)ga;                         // bits 95:64  global_addr[31:0]
  g0[3] = ((unsigned)(ga >> 32) & 0x01FFFFFFu)  // bits 120:96 global_addr[56:32]
          | (2u << 30);                         // bits 127:126 type=2 ("image")
  i32x8 g1;
  g1[0] = (int)(2u << 16);                      // data_size=2 (4 bytes); mask=0
  g1[1] = (int)((unsigned)NDIM << 16);          // tensor_dim0[15:0] @ bits 63:48
  g1[2] = (int)((unsigned)NDIM << 16);          // dim0[31:16]=0 | tensor_dim1[15:0]
  g1[3] = (int)((unsigned)MTILE << 16);         // dim1[31:16]=0 | tile_dim0=32
  g1[4] = (int)(unsigned)MTILE;                 // tile_dim1=32 | tile_dim2=0
  g1[5] = NDIM;                                 // tensor_dim0_stride[31:0]=384
  g1[6] = 0;                                    // stride[47:32]=0 | dim1_stride lo
  g1[7] = 0;
  i32x4 z4 = {0, 0, 0, 0};                      // groups 2/3 unused (2D tensor)
  i32x8 z8 = {0, 0, 0, 0, 0, 0, 0, 0};
  __builtin_amdgcn_tensor_load_to_lds(g0, g1, z4, z4, z8, 0);
}

// ---------------------------------------------------------------------------
// Twiddle matrix: F[j][k] = exp(-2*pi*i * j*k / N)  (symmetric)
// ---------------------------------------------------------------------------
__global__ void k_twiddle(float* __restrict__ Fr, float* __restrict__ Fi) {
  int idx = blockIdx.x * blockDim.x + threadIdx.x;  // NN total
  int j = idx / NDIM;
  int k = idx % NDIM;
  int t = (j * k) % NDIM;  // reduce mod N for precision
  float ang = -2.0f * 3.14159265358979323846f * (float)t / (float)NDIM;
  float s, c;
  sincosf(ang, &s, &c);
  Fr[idx] = c;
  Fi[idx] = s;
}

// ---------------------------------------------------------------------------
// Sum of running_amp (to evaluate the jnp.where(sum == 0, ...) branch)
// ---------------------------------------------------------------------------
__global__ void k_sum(const float* __restrict__ in, float* __restrict__ out,
                      int n) {
  __shared__ float sm[256];
  float acc = 0.f;
  for (int i = blockIdx.x * blockDim.x + threadIdx.x; i < n;
       i += gridDim.x * blockDim.x)
    acc += in[i];
  sm[threadIdx.x] = acc;
  __syncthreads();
  for (int s = 128; s > 0; s >>= 1) {
    if ((int)threadIdx.x < s) sm[threadIdx.x] += sm[threadIdx.x + s];
    __syncthreads();
  }
  if (threadIdx.x == 0) atomicAdd(out, sm[0]);
}

// ---------------------------------------------------------------------------
// Generalized complex GEMM stage, TDM-fed, double-buffered LDS, WMMA compute.
//   C = op(A) * op(B) accumulated over K=384, all matrices 384x384 (ld=384).
// MODE 0: A complex (F), B real        -> Cr = Ar*B       , Ci = Ai*B
// MODE 1: A complex,     B complex     -> C  = A * B       (forward col)
// MODE 2: A complex conj,B complex     -> C  = conj(A) * B (inverse row)
// MODE 3: A complex,     B complex conj-> Cr = real(A*conj(B)) only (inv col)
//         (real part = Ar*Br + Ai*Bi since conj(B) flips Bi sign)
// ---------------------------------------------------------------------------
template <int MODE>
__global__ void __launch_bounds__(128) k_cgemm(
    const float* __restrict__ Apr, const float* __restrict__ Api,
    const float* __restrict__ Bpr, const float* __restrict__ Bpi,
    float* __restrict__ Cpr, float* __restrict__ Cpi, long strideA,
    long strideB, float scale) {
  __shared__ float smem[2][4][MTILE * MTILE];  // [buf][slab Ar,Ai,Br,Bi] 32KB

  const int lane = threadIdx.x & 31;
  const int wave = threadIdx.x >> 5;
  const int n = lane & 15;
  const int hi = lane >> 4;

  const int plane = blockIdx.x / TILES_PER_PLANE;
  const int t = blockIdx.x % TILES_PER_PLANE;
  const int rowB = (t / NMT) * MTILE;
  const int colB = (t % NMT) * MTILE;

  const float* Ar = Apr + (size_t)plane * strideA;
  const float* Ai = Api + (size_t)plane * strideA;
  const float* Br = Bpr + (size_t)plane * strideB;
  const float* Bi = (MODE != 0) ? (Bpi + (size_t)plane * strideB) : nullptr;

  // One TDM tile per wave per chunk; TDM ignores EXEC, branch is wave-uniform.
  auto issue = [&](int kb, int buf) {
    if (wave == 0)
      tdm_load_tile32(Ar + (size_t)rowB * NDIM + kb,
                      (unsigned)(uintptr_t)&smem[buf][0][0]);
    else if (wave == 1)
      tdm_load_tile32(Ai + (size_t)rowB * NDIM + kb,
                      (unsigned)(uintptr_t)&smem[buf][1][0]);
    else if (wave == 2)
      tdm_load_tile32(Br + (size_t)kb * NDIM + colB,
                      (unsigned)(uintptr_t)&smem[buf][2][0]);
    else if (MODE != 0 && wave == 3)
      tdm_load_tile32(Bi + (size_t)kb * NDIM + colB,
                      (unsigned)(uintptr_t)&smem[buf][3][0]);
  };

  v8f p_rr = {0.f, 0.f, 0.f, 0.f, 0.f, 0.f, 0.f, 0.f};
  v8f p_ir = p_rr, p_ri = p_rr, p_ii = p_rr;

  // Row of A fragment / col of B fragment inside the 32x32 macro tile.
  const int am = (wave >> 1) * 16 + n;
  const int bn = (wave & 1) * 16 + n;

  issue(0, 0);
  __builtin_amdgcn_s_wait_tensorcnt(0);
  __syncthreads();

  for (int c = 0; c < NCHUNK; ++c) {
    const int buf = c & 1;
    if (c + 1 < NCHUNK) issue((c + 1) * KC, buf ^ 1);  // prefetch next chunk

    const float* As_r = &smem[buf][0][0];
    const float* As_i = &smem[buf][1][0];
    const float* Bs_r = &smem[buf][2][0];
    const float* Bs_i = &smem[buf][3][0];

#pragma unroll
    for (int kk = 0; kk < KC; kk += 4) {
      const int ka = kk + 2 * hi;  // lane-half selects K pair (ISA §7.12.2)
      v2f ar, ai, br, bi;
      ar.x = As_r[am * MTILE + ka];
      ar.y = As_r[am * MTILE + ka + 1];
      ai.x = As_i[am * MTILE + ka];
      ai.y = As_i[am * MTILE + ka + 1];
      br.x = Bs_r[ka * MTILE + bn];
      br.y = Bs_r[(ka + 1) * MTILE + bn];
      p_rr = wmma4(ar, br, p_rr);
      if constexpr (MODE != 3) p_ir = wmma4(ai, br, p_ir);
      if constexpr (MODE != 0) {
        bi.x = Bs_i[ka * MTILE + bn];
        bi.y = Bs_i[(ka + 1) * MTILE + bn];
        if constexpr (MODE == 1 || MODE == 2) p_ri = wmma4(ar, bi, p_ri);
        p_ii = wmma4(ai, bi, p_ii);
      }
    }
    __builtin_amdgcn_s_wait_tensorcnt(0);  // prefetch landed
    __syncthreads();                        // all waves may flip buffers
  }

  const int outRow = rowB + (wave >> 1) * 16;
  const int outCol = colB + (wave & 1) * 16;
  v8f cr = p_rr, ci = p_ir;
  if constexpr (MODE == 1) {
    cr = p_rr - p_ii;
    ci = p_ri + p_ir;
  } else if constexpr (MODE == 2) {
    cr = p_rr + p_ii;  // conj(A): (Ar - iAi)(Br + iBi)
    ci = p_ri - p_ir;
  } else if constexpr (MODE == 3) {
    cr = p_rr + p_ii;  // real(A * conj(B))
  }
  store_c(Cpr + (size_t)plane * NN, outRow, outCol, lane, cr, scale);
  if constexpr (MODE != 3)
    store_c(Cpi + (size_t)plane * NN, outRow, outCol, lane, ci, scale);
}

// ---------------------------------------------------------------------------
// Batch-mean amplitude + EMA with running_amp
// ---------------------------------------------------------------------------
__global__ void k_amp(const float* __restrict__ Zr, const float* __restrict__ Zi,
                      const float* __restrict__ running,
                      const float* __restrict__ sumflag,
                      float* __restrict__ amp) {
  int idx = blockIdx.x * blockDim.x + threadIdx.x;  // 3*NN total
  int c = idx / NN;
  int r = idx % NN;
  float s = 0.f;
#pragma unroll 4
  for (int b = 0; b < 32; ++b) {
    size_t o = (size_t)(b * 3 + c) * NN + r;
    float zr = Zr[o], zi = Zi[o];
    s += sqrtf(zr * zr + zi * zi);
  }
  float m = s * (1.0f / 32.0f);
  float prev = running[idx];
  amp[idx] = (*sumflag == 0.0f) ? m : (1.0f - MOMENTUM) * prev + MOMENTUM * m;
}

// ---------------------------------------------------------------------------
// Rebuild spectrum in place:  W = amp * Z / |Z|   (amp broadcast over batch)
// ---------------------------------------------------------------------------
__global__ void k_phase(float* __restrict__ Zr, float* __restrict__ Zi,
                        const float* __restrict__ amp) {
  int idx = blockIdx.x * blockDim.x + threadIdx.x;  // NPLANES*NN total
  int p = idx / NN;
  int r = idx % NN;
  int c = p % 3;
  float a = amp[(size_t)c * NN + r];
  size_t o = (size_t)idx;
  float zr = Zr[o], zi = Zi[o];
  float mag = sqrtf(zr * zr + zi * zi);
  if (mag > 0.f) {
    float sc = a / mag;
    Zr[o] = zr * sc;
    Zi[o] = zi * sc;
  } else {
    Zr[o] = a;  // angle(0)==0 -> exp(i*0)==1
    Zi[o] = 0.f;
  }
}

// ---------------------------------------------------------------------------
extern "C" void kernel_launch(void* const* d_in, const int* in_sizes, int n_in,
                              void* d_out, int out_size, void* d_ws,
                              size_t ws_size, hipStream_t stream) {
  const float* x = (const float*)d_in[0];        // [32,3,384,384]
  const float* running = (const float*)d_in[1];  // [3,384,384]
  float* out = (float*)d_out;                    // [32,3,384,384]

  float* ws = (float*)d_ws;
  const size_t nn = (size_t)NN;
  float* Fr = ws;                          // NN
  float* Fi = Fr + nn;                     // NN
  float* Yr = Fi + nn;                     // NPLANES*NN (also U)
  float* Yi = Yr + (size_t)NPLANES * nn;
  float* Zr = Yi + (size_t)NPLANES * nn;   // NPLANES*NN (also W)
  float* Zi = Zr + (size_t)NPLANES * nn;
  float* amp = Zi + (size_t)NPLANES * nn;  // 3*NN
  float* sumbuf = amp + 3 * nn;            // 1

  const int gemmBlocks = NPLANES * TILES_PER_PLANE;  // 13824 WGs x 128 threads

  k_twiddle<<<NN / 256, 256, 0, stream>>>(Fr, Fi);
  (void)hipMemsetAsync(sumbuf, 0, sizeof(float), stream);
  k_sum<<<432, 256, 0, stream>>>(running, sumbuf, 3 * NN);

  // Forward:  Y = F * x ;  Z = Y * F
  k_cgemm<0><<<gemmBlocks, 128, 0, stream>>>(Fr, Fi, x, nullptr, Yr, Yi,
                                             0L, (long)NN, 1.0f);
  k_cgemm<1><<<gemmBlocks, 128, 0, stream>>>(Yr, Yi, Fr, Fi, Zr, Zi,
                                             (long)NN, 0L, 1.0f);

  k_amp<<<(3 * NN) / 256, 256, 0, stream>>>(Zr, Zi, running, sumbuf, amp);
  k_phase<<<(NPLANES * NN) / 256, 256, 0, stream>>>(Zr, Zi, amp);

  // Inverse:  U = conj(F) * W ;  out = real(U * conj(F)) / N^2
  k_cgemm<2><<<gemmBlocks, 128, 0, stream>>>(Fr, Fi, Zr, Zi, Yr, Yi,
                                             0L, (long)NN, 1.0f);
  k_cgemm<3><<<gemmBlocks, 128, 0, stream>>>(Yr, Yi, Fr, Fi, out, nullptr,
                                             (long)NN, 0L, 1.0f / (float)NN);
}